// ReprojectionLayer_83468394431049
// MI455X (gfx1250) — compile-verified
//
#include <hip/hip_runtime.h>
#include <hip/hip_bf16.h>

// ReprojectionLayer for MI455X (gfx1250, wave32).
//
// See round-1 notes: one wave handles 16 consecutive voxels; the 12-camera
// projection of those voxels is computed with two V_WMMA_F32_16X16X4_F32 ops
// (A = camera coefficient rows: cams 0..5 in rows 0..5, cams 6..11 in rows
// 8..13; B = voxel homogeneous coords as columns).  The D-matrix layout puts
// cams 0..5 of voxel L in lane L and cams 6..11 in lane L+16 -> each lane
// gathers 6 cams x 16 joints, shfl_xor(16) folds the halves, lanes 0..15
// write 16 joints/voxel with non-temporal stores (protect the 192MB L2,
// which nearly holds the 252MB heatmap working set, from the 57MB output
// stream).
//
// Round-2 change: replace the 12 IEEE-accurate f32 divisions per lane
// (v_div_scale/v_div_fmas expansions dominated the VALU count) with one fast
// v_rcp_f32 of the voxel z and 12 multiplies.  The result only feeds
// clamp + truncate-to-2px-bin, so ~1-ulp rcp accuracy is ample.

#define G     96
#define G3    (96 * 96 * 96)
#define NJ    16
#define HH    512
#define WW    640
#define HW    (512 * 640)
#define NCAM  12

typedef float v2f __attribute__((ext_vector_type(2)));
typedef float v8f __attribute__((ext_vector_type(8)));

__global__ __launch_bounds__(256)
void ReprojectionLayer_83468394431049_kernel(const float* __restrict__ hm,      // (1, 12, 16, 512, 640)
                                             const float* __restrict__ center,  // (3,)
                                             const float* __restrict__ cm,      // (12, 4, 3)
                                             float* __restrict__ out) {         // (1, 16, 96, 96, 96)
  const int lane = threadIdx.x & 31;
  const int wave = threadIdx.x >> 5;
  const int m    = lane & 15;   // A-row index / D-column (voxel-in-group) index
  const int khi  = lane >> 4;   // 0 -> K0,K1 half ; 1 -> K2,K3 half

  const int voxbase = (blockIdx.x * 8 + wave) * 16;
  const int vox = voxbase + m;  // this lane's voxel (flat, z-fastest)

  // --- voxel world coordinates ---------------------------------------------
  int iz = vox % G;
  int t  = vox / G;
  int iy = t % G;
  int ix = t / G;
  const float cx = center[0], cy = center[1], cz = center[2];
  const float xc = (float)((ix < G / 2) ? ix : ix - G) * 2.0f + cx;
  const float yc = (float)((iy < G / 2) ? iy : iy - G) * 2.0f + cy;
  const float zc = (float)((iz < G / 2) ? iz : iz - G) * 2.0f + cz;

  // --- build A (camera coefficient rows) for u and v projections ------------
  // proj[c, p] = sum_k xh[k] * cm[c*12 + k*3 + p]
  const int cam_row = (m < 6) ? m : ((m >= 8 && m < 14) ? (m - 2) : -1);
  const int k0 = 2 * khi, k1 = 2 * khi + 1;
  v2f aU = {0.0f, 0.0f};
  v2f aV = {0.0f, 0.0f};
  if (cam_row >= 0) {
    const float* P = cm + cam_row * 12;
    aU.x = P[k0 * 3 + 0];
    aU.y = P[k1 * 3 + 0];
    aV.x = P[k0 * 3 + 1];
    aV.y = P[k1 * 3 + 1];
  }

  // --- build B (voxel homogeneous coords as columns) -------------------------
  v2f b;
  b.x = khi ? zc : xc;   // rows: K0 = x, K1 = y, K2 = z, K3 = 1
  b.y = khi ? 1.0f : yc;

  v8f zero8 = {0.f, 0.f, 0.f, 0.f, 0.f, 0.f, 0.f, 0.f};
  // D = A x B : two WMMAs give u_raw and v_raw for all 12 cams x 16 voxels.
  v8f du = __builtin_amdgcn_wmma_f32_16x16x4_f32(false, aU, false, b,
                                                 (short)0, zero8, false, false);
  v8f dv = __builtin_amdgcn_wmma_f32_16x16x4_f32(false, aV, false, b,
                                                 (short)0, zero8, false, false);
  // This lane now holds du[c], dv[c] for cams (khi ? 6..11 : 0..5) of voxel m.

  // --- gather + accumulate over this lane's 6 cameras, all 16 joints ---------
  float acc[NJ];
#pragma unroll
  for (int j = 0; j < NJ; ++j) acc[j] = 0.0f;

  // One fast hardware reciprocal (v_rcp_f32) instead of 12 IEEE divisions:
  // z in [504, 695], and the quotient only feeds clamp + 2px truncation.
  const float rz = __builtin_amdgcn_rcpf(zc);

  const int cambase = khi * 6;
#pragma unroll
  for (int cc = 0; cc < 6; ++cc) {
    float u = du[cc] * rz;
    float v = dv[cc] * rz;
    u = fminf(fmaxf(u, 0.0f), 1279.0f);
    v = fminf(fmaxf(v, 0.0f), 1023.0f);
    const int idx = (int)(v * 0.5f) * WW + (int)(u * 0.5f);
    const float* base = hm + (size_t)(cambase + cc) * (NJ * HW) + idx;
#pragma unroll
    for (int j = 0; j < NJ; ++j) {
      acc[j] += base[(size_t)j * HW];   // default RT temporal hint: stay in L2
    }
  }

  // --- fold cams 6..11 (upper half-wave) into cams 0..5 (lower half) ---------
#pragma unroll
  for (int j = 0; j < NJ; ++j) {
    acc[j] += __shfl_xor(acc[j], 16, 32);   // wave32
  }

  // --- lanes 0..15 write 16 joints for their voxel (coalesced over lanes) ----
  if (lane < 16) {
    const float scale = 1.0f / (float)NCAM;
#pragma unroll
    for (int j = 0; j < NJ; ++j) {
      __builtin_nontemporal_store(acc[j] * scale, out + (size_t)j * G3 + vox);
    }
  }
}

extern "C" void kernel_launch(void* const* d_in, const int* in_sizes, int n_in,
                              void* d_out, int out_size, void* d_ws, size_t ws_size,
                              hipStream_t stream) {
  const float* heatmaps = (const float*)d_in[0];  // 1*12*16*512*640 f32
  const float* center   = (const float*)d_in[1];  // 3 f32
  const float* cammats  = (const float*)d_in[2];  // 12*4*3 f32
  float* out            = (float*)d_out;          // 1*16*96*96*96 f32

  // 8 waves/block * 16 voxels/wave = 128 voxels/block; 96^3 / 128 = 6912 blocks.
  const int blocks = G3 / 128;
  ReprojectionLayer_83468394431049_kernel<<<blocks, 256, 0, stream>>>(
      heatmaps, center, cammats, out);
}